// YOLOV3Loss_45483703664642
// MI455X (gfx1250) — compile-verified
//
#include <hip/hip_runtime.h>
#include <stdint.h>

// Problem constants (fixed by setup_inputs in the reference)
#define B_IMG   32
#define NA      3
#define GSZ     52
#define NC      80
#define NM      50
#define A_TOT   (NA * GSZ * GSZ)      // 8112 anchors per image
#define INV_S   0.125f                // 1/stride, stride=8
#define EPSF    1e-7f

typedef float v2f __attribute__((ext_vector_type(2)));
typedef float v8f __attribute__((ext_vector_type(8)));

// ---------------------------------------------------------------------------
// Wave32 partial reduction via V_WMMA_F32_16X16X4_F32 with B = ones:
// D[m][n] = sum_k A[m][k], so summing the 8 D VGPRs gives, in lanes 0-15,
// the sum of rows 0..7 and, in lanes 16-31, the sum of rows 8..15.  The two
// half-sums together equal the 32-lane total (robust to A's K-slot packing
// since B is all ones).  Caller combines the two halves through LDS.
// EXEC must be all ones: every thread of the (full) block calls this.
// ---------------------------------------------------------------------------
__device__ __forceinline__ float wave_halfsum_wmma(float x) {
#if __has_builtin(__builtin_amdgcn_wmma_f32_16x16x4_f32)
  v2f a; a[0] = x;    a[1] = 0.0f;
  v2f b; b[0] = 1.0f; b[1] = 1.0f;
  v8f c = {};
  v8f d = __builtin_amdgcn_wmma_f32_16x16x4_f32(
      /*neg_a=*/false, a, /*neg_b=*/false, b,
      /*c_mod=*/(short)0, c, /*reuse_a=*/false, /*reuse_b=*/false);
  return d[0] + d[1] + d[2] + d[3] + d[4] + d[5] + d[6] + d[7];
#else
  // fallback: full butterfly; both half-lanes then hold the same total,
  // which still sums correctly (caller divides work across 2 slots).
  for (int o = 16; o > 0; o >>= 1) x += __shfl_xor(x, o, 32);
  return 0.5f * x;
#endif
}

__device__ __forceinline__ float sigmoidf_(float x) {
  return 1.0f / (1.0f + __expf(-x));
}
__device__ __forceinline__ float clip01(float v) {
  return fminf(fmaxf(v, EPSF), 1.0f - EPSF);
}

__global__ void yolo_zero_kernel(float* out, int n) {
  int i = blockIdx.x * blockDim.x + threadIdx.x;
  if (i < n) out[i] = 0.0f;
}

__global__ __launch_bounds__(256)
void yolo_loss_kernel(const float* __restrict__ pred,
                      const float* __restrict__ bbox,
                      const float* __restrict__ anch,
                      float* __restrict__ out) {
  __shared__ float s_raw[NM * 5];
  __shared__ float s_bx1[NM], s_by1[NM], s_bx2[NM], s_by2[NM], s_area[NM];
  __shared__ float s_gcx[NM], s_gcy[NM], s_gw[NM], s_gh[NM], s_cls[NM];
  __shared__ int   s_valid[NM];
  __shared__ int   s_has;
  __shared__ float s_part[3][16];    // 2 half-sums per wave, 8 waves

  const int tid = threadIdx.x;
  const int img = blockIdx.x;

  // ---- CDNA5 async copy: broadcast this image's 50 GT boxes (1000B) to LDS
  {
    const float* gsrc = bbox + (size_t)img * (NM * 5);
    if (tid < NM * 5) {
      unsigned ldsa = (unsigned)(uintptr_t)(&s_raw[tid]);  // low 32b = LDS offset
      unsigned voff = (unsigned)(tid * 4);
      asm volatile("global_load_async_to_lds_b32 %0, %1, %2"
                   :: "v"(ldsa), "v"(voff), "s"(gsrc) : "memory");
    }
    asm volatile("s_wait_asynccnt 0" ::: "memory");
  }
  __syncthreads();

  // ---- derive per-GT quantities once per block
  if (tid < NM) {
    float x  = s_raw[tid * 5 + 0];
    float y  = s_raw[tid * 5 + 1];
    float w  = s_raw[tid * 5 + 2];
    float h  = s_raw[tid * 5 + 3];
    float cl = s_raw[tid * 5 + 4];
    s_bx1[tid]  = x;                 // cx - w/2 == x
    s_by1[tid]  = y;
    s_bx2[tid]  = x + w;
    s_by2[tid]  = y + h;
    s_area[tid] = w * h;
    s_gcx[tid]  = x + 0.5f * w;
    s_gcy[tid]  = y + 0.5f * h;
    s_gw[tid]   = w;
    s_gh[tid]   = h;
    s_cls[tid]  = cl;
    s_valid[tid] = (cl != -1.0f) ? 1 : 0;
  }
  if (tid == 0) {
    int h = 0;
    for (int m = 0; m < NM; ++m) h |= (s_raw[m * 5 + 4] != -1.0f) ? 1 : 0;
    s_has = h;
  }
  __syncthreads();

  const int a = blockIdx.y * blockDim.x + tid;

  float accC = 0.0f, accR = 0.0f, accO = 0.0f;

  if (a < A_TOT) {
    const float* row = pred + ((size_t)img * A_TOT + (size_t)a) * (NC + 5);
    // prefetch the 340B prediction row while we do the IoU loop
    __builtin_prefetch(row,      0, 3);
    __builtin_prefetch(row + 32, 0, 3);
    __builtin_prefetch(row + 64, 0, 3);

    const float acx = anch[(size_t)a * 4 + 0];
    const float acy = anch[(size_t)a * 4 + 1];
    const float aw  = anch[(size_t)a * 4 + 2];
    const float ah  = anch[(size_t)a * 4 + 3];
    const float ax1 = acx - 0.5f * aw, ax2 = acx + 0.5f * aw;
    const float ay1 = acy - 0.5f * ah, ay2 = acy + 0.5f * ah;
    const float areaA = aw * ah;

    // per-anchor IoU max / argmax over 50 GT boxes (strict > keeps first max)
    float imax = -1.0f;
    int   iarg = 0;
    #pragma unroll 5
    for (int m = 0; m < NM; ++m) {
      float iw = fmaxf(fminf(ax2, s_bx2[m]) - fmaxf(ax1, s_bx1[m]), 0.0f);
      float ih = fmaxf(fminf(ay2, s_by2[m]) - fmaxf(ay1, s_by1[m]), 0.0f);
      float inter = iw * ih;
      float iou = inter / (areaA + s_area[m] - inter + 1e-16f);
      iou = s_valid[m] ? iou : -1.0f;
      if (iou > imax) { imax = iou; iarg = m; }
    }
    const bool pos = (imax >= 0.5f);

    // confidence MSE: obj weight 1.0, noobj weight 0.5
    {
      float cf = row[4];
      accO = pos ? (cf - 1.0f) * (cf - 1.0f) : 0.5f * cf * cf;
    }

    // class BCE:  -sum log1p(-p)  plus per-positive correction on the target
    {
      float s = 0.0f;
      #pragma unroll 8
      for (int c = 0; c < NC; ++c) {
        float pc = clip01(row[5 + c]);
        s -= log1pf(-pc);
      }
      if (pos) {
        int   ci = (int)s_cls[iarg];
        float pc = clip01(row[5 + ci]);
        s += log1pf(-pc) - __logf(pc);   // swap the (1-y) term for the y term
      }
      accC = s;
    }

    // regression on positives
    if (pos) {
      float gx = s_gcx[iarg] * INV_S, gy = s_gcy[iarg] * INV_S;
      float gw = fmaxf(s_gw[iarg], 1.0f), gh = fmaxf(s_gh[iarg], 1.0f);
      float stx  = sigmoidf_(row[0]);
      float sty  = sigmoidf_(row[1]);
      float stxg = sigmoidf_(gx - acx * INV_S);
      float styg = sigmoidf_(gy - acy * INV_S);
      float twg = __logf(gw / aw + 1e-16f);
      float thg = __logf(gh / ah + 1e-16f);
      float param = 2.0f - fabsf(twg) * fabsf(thg);
      float dx = stx - stxg, dy = sty - styg;
      float dw = row[2] - twg, dh = row[3] - thg;
      accR = 5.0f * param * (dx * dx + dy * dy + dw * dw + dh * dh);
    }
  }

  // has-any-valid gating (uniform per block)
  if (!s_has) { accC = 0.0f; accR = 0.0f; accO = 0.0f; }

  // ---- wave reduction via WMMA (all 256 threads execute: EXEC all 1s)
  float tC = wave_halfsum_wmma(accC);
  float tR = wave_halfsum_wmma(accR);
  float tO = wave_halfsum_wmma(accO);

  // lanes 0 and 16 of each wave hold the two half-sums; no cross-lane
  // shuffle needed — both write straight to LDS.
  if ((tid & 15) == 0) {
    const int slot = tid >> 4;     // 0..15 : (wave, half)
    s_part[0][slot] = tC;
    s_part[1][slot] = tR;
    s_part[2][slot] = tO;
  }
  __syncthreads();

  if (tid == 0) {
    float c = 0.0f, r = 0.0f, o = 0.0f;
    #pragma unroll
    for (int w = 0; w < 16; ++w) {
      c += s_part[0][w];
      r += s_part[1][w];
      o += s_part[2][w];
    }
    const float invB = 1.0f / (float)B_IMG;   // mean over images
    atomicAdd(&out[0], c * invB);
    atomicAdd(&out[1], r * invB);
    atomicAdd(&out[2], o * invB);
  }
}

extern "C" void kernel_launch(void* const* d_in, const int* in_sizes, int n_in,
                              void* d_out, int out_size, void* d_ws, size_t ws_size,
                              hipStream_t stream) {
  (void)in_sizes; (void)n_in; (void)d_ws; (void)ws_size;
  const float* pred = (const float*)d_in[0];   // (B, NA, G, G, 85)
  const float* bbox = (const float*)d_in[1];   // (B, 50, 5)
  const float* anch = (const float*)d_in[2];   // (NA, G, G, 4)
  float* out = (float*)d_out;                  // [cls_mean, reg_mean, conf_mean]

  yolo_zero_kernel<<<1, 32, 0, stream>>>(out, out_size);

  dim3 grid(B_IMG, (A_TOT + 255) / 256);
  yolo_loss_kernel<<<grid, 256, 0, stream>>>(pred, bbox, anch, out);
}